// MovementPruner_29291676958791
// MI455X (gfx1250) — compile-verified
//
#include <hip/hip_runtime.h>

typedef unsigned int uint32;

#define TILE 4096              // floats per TDM tile (16 KB)
#define HIST_BLOCKS 1024
#define SCAN_THREADS 256

// ---- workspace layout (uint32 words) ----
//  [0    .. 4095]  hist level0 (4096 bins, key[31:20])
//  [4096 .. 8191]  hist level1 (4096 bins, key[19:8])
//  [8192 .. 8447]  hist level2 ( 256 bins, key[7:0])
//  [8448]          prefix (selected high bits so far)
//  [8449]          k remaining within current prefix
//  [8450]          threshold float bits
#define STATE_OFF 8448
#define WS_WORDS  8451

// Monotonic order-preserving remap: larger float -> larger unsigned key.
__device__ __forceinline__ uint32 mono_key(float s) {
  uint32 u = __float_as_uint(s);
  return (u & 0x80000000u) ? ~u : (u | 0x80000000u);
}

__global__ void init_ws(uint32* __restrict__ ws, uint32 k) {
  int i = blockIdx.x * blockDim.x + threadIdx.x;
  if (i < WS_WORDS) ws[i] = (i == STATE_OFF + 1) ? k : 0u;
}

template <int LEVEL>
__device__ __forceinline__ void hist_update(uint32* __restrict__ hist, uint32 prefix, float s) {
  uint32 key = mono_key(s);
  if (LEVEL == 1) {
    if ((key >> 20) == prefix) atomicAdd(&hist[(key >> 8) & 0xFFFu], 1u);
  } else {
    if ((key >> 8) == prefix) atomicAdd(&hist[key & 0xFFu], 1u);
  }
}

template <int LEVEL>
__global__ void hist_kernel(const float4* __restrict__ s4, long long n4,
                            uint32* __restrict__ hist, const uint32* __restrict__ state) {
  long long stride = (long long)gridDim.x * blockDim.x;
  long long i0 = (long long)blockIdx.x * blockDim.x + threadIdx.x;
  if (LEVEL == 0) {
    __shared__ uint32 lh[4096];
    for (int i = threadIdx.x; i < 4096; i += blockDim.x) lh[i] = 0u;
    __syncthreads();
    for (long long i = i0; i < n4; i += stride) {
      float4 v = s4[i];
      atomicAdd(&lh[mono_key(v.x) >> 20], 1u);
      atomicAdd(&lh[mono_key(v.y) >> 20], 1u);
      atomicAdd(&lh[mono_key(v.z) >> 20], 1u);
      atomicAdd(&lh[mono_key(v.w) >> 20], 1u);
    }
    __syncthreads();
    for (int i = threadIdx.x; i < 4096; i += blockDim.x) {
      uint32 c = lh[i];
      if (c) atomicAdd(&hist[i], c);
    }
  } else {
    uint32 prefix = state[0];  // scores are L2-resident (134 MB < 192 MB L2)
    for (long long i = i0; i < n4; i += stride) {
      float4 v = s4[i];
      hist_update<LEVEL>(hist, prefix, v.x);
      hist_update<LEVEL>(hist, prefix, v.y);
      hist_update<LEVEL>(hist, prefix, v.z);
      hist_update<LEVEL>(hist, prefix, v.w);
    }
  }
}

// Single-block scan: descend from the top bin to find the bin containing the
// k-th largest key; update (prefix, k_remaining); optionally finalize threshold.
__global__ void scan_kernel(const uint32* __restrict__ hist, uint32* __restrict__ state,
                            int nbins, int newbits, int finalize) {
  __shared__ uint32 hl[4096];
  __shared__ uint32 partial[SCAN_THREADS];
  int t = threadIdx.x;
  for (int i = t; i < nbins; i += blockDim.x) hl[i] = hist[i];
  __syncthreads();
  int chunk = nbins / SCAN_THREADS;  // 16 or 1
  uint32 sum = 0;
  for (int i = 0; i < chunk; ++i) sum += hl[t * chunk + i];
  partial[t] = sum;
  __syncthreads();
  if (t == 0) {
    uint32 k = state[1];
    uint32 cum = 0;
    int c;
    for (c = SCAN_THREADS - 1; c > 0; --c) {
      if (cum + partial[c] >= k) break;
      cum += partial[c];
    }
    int lo = c * chunk, b;
    for (b = lo + chunk - 1; b > lo; --b) {
      uint32 h = hl[b];
      if (cum + h >= k) break;
      cum += h;
    }
    uint32 prefix = (state[0] << newbits) | (uint32)b;
    state[0] = prefix;
    state[1] = k - cum;
    if (finalize) {
      uint32 key = prefix;  // full 32-bit key of the k-th largest score
      state[2] = (key & 0x80000000u) ? (key ^ 0x80000000u) : ~key;
    }
  }
}

// ---------------- CDNA5 Tensor Data Mover path ----------------
#if defined(__AMDGCN__) && __has_builtin(__builtin_amdgcn_tensor_load_to_lds)
#define TDM_OK 1
#else
#define TDM_OK 0
#endif

__device__ __forceinline__ void wait_tensorcnt0() {
#if defined(__AMDGCN__) && __has_builtin(__builtin_amdgcn_s_wait_tensorcnt)
  __builtin_amdgcn_s_wait_tensorcnt(0);
#else
  asm volatile("s_wait_tensorcnt 0x0" ::: "memory");
#endif
}
__device__ __forceinline__ void wait_tensorcnt2() {
#if defined(__AMDGCN__) && __has_builtin(__builtin_amdgcn_s_wait_tensorcnt)
  __builtin_amdgcn_s_wait_tensorcnt(2);
#else
  asm volatile("s_wait_tensorcnt 0x2" ::: "memory");
#endif
}

#if TDM_OK
typedef unsigned int v4u __attribute__((ext_vector_type(4)));
typedef int v8i __attribute__((ext_vector_type(8)));
typedef int v4i __attribute__((ext_vector_type(4)));

// 1-row tile load: `elems` fp32 from gptr -> LDS byte offset lds_byte.
// D# per CDNA5 ISA ch.8: group0 = {count=1 | lds_addr | global_addr | type=2},
// group1 = {data_size=4B, tensor_dim0=elems, tensor_dim1=1, tile_dim0=elems,
//           tile_dim1=1, stride0=elems, stride1=elems}. Groups 2/3 zero (2D);
// trailing v8i is the toolchain's extra group (zero-filled, probe-verified).
__device__ __forceinline__ void tdm_load_1d(const void* gptr, uint32 lds_byte, uint32 elems) {
  unsigned long long ga = (unsigned long long)gptr;
  v4u g0;
  g0[0] = 1u;                                                   // count=1 (user D#)
  g0[1] = lds_byte;                                             // lds_addr (bytes)
  g0[2] = (uint32)(ga & 0xFFFFFFFFull);                         // global_addr[31:0]
  g0[3] = (uint32)((ga >> 32) & 0x01FFFFFFull) | (2u << 30);    // addr[56:32] | type=2
  v8i g1;
  g1[0] = (int)(2u << 16);                                      // data_size = 4 bytes
  g1[1] = (int)((elems & 0xFFFFu) << 16);                       // tensor_dim0[15:0]
  g1[2] = (int)((elems >> 16) | (1u << 16));                    // tensor_dim0[31:16], tensor_dim1=1
  g1[3] = (int)((elems & 0xFFFFu) << 16);                       // tile_dim0
  g1[4] = 1;                                                    // tile_dim1=1, tile_dim2=0
  g1[5] = (int)elems;                                           // tensor_dim0_stride[31:0]
  g1[6] = (int)((elems & 0xFFFFu) << 16);                       // stride0 hi=0, stride1[15:0]
  g1[7] = 0;                                                    // stride1 hi
  v4i gz4 = {0, 0, 0, 0};
  v8i gz8 = {0, 0, 0, 0, 0, 0, 0, 0};
  __builtin_amdgcn_tensor_load_to_lds(g0, g1, gz4, gz4, gz8, 0);
}
#endif

__global__ void mask_kernel(const float* __restrict__ scores, const float* __restrict__ x,
                            float* __restrict__ out, const uint32* __restrict__ state,
                            long long tiles_total, int tiles_per_block) {
  float thresh = __uint_as_float(state[2]);
#if TDM_OK
  // Double-buffered TDM pipeline: wave 0 DMAs tile t+1 (scores,x) into LDS
  // while all 8 waves compute/store tile t.  64 KB static LDS, 4 x 16 KB bufs.
  __shared__ float smem[4 * TILE];
  long long base = (long long)blockIdx.x * tiles_per_block;
  long long nt = tiles_total - base;
  if (nt <= 0) return;
  if (nt > tiles_per_block) nt = tiles_per_block;
  const bool issuer = (threadIdx.x < 32);  // wave 0 only (EXEC ignored by TDM)
  if (issuer) {
    tdm_load_1d(scores + base * TILE, 0u, TILE);
    tdm_load_1d(x + base * TILE, (uint32)(TILE * sizeof(float)), TILE);
  }
  for (int i = 0; i < (int)nt; ++i) {
    int pair = i & 1;
    if (issuer) {
      if (i + 1 < (int)nt) {
        long long t = base + i + 1;
        uint32 off = (uint32)(((i + 1) & 1) * 2 * TILE * sizeof(float));
        tdm_load_1d(scores + t * TILE, off, TILE);
        tdm_load_1d(x + t * TILE, off + (uint32)(TILE * sizeof(float)), TILE);
        wait_tensorcnt2();   // tile i complete; tile i+1 (2 ops) in flight
      } else {
        wait_tensorcnt0();
      }
    }
    __syncthreads();  // publish LDS tile i to all waves
    const float4* s4 = (const float4*)&smem[pair * 2 * TILE];
    const float4* x4 = (const float4*)&smem[pair * 2 * TILE + TILE];
    float4* o4 = (float4*)(out + (base + i) * TILE);
    for (int j = threadIdx.x; j < TILE / 4; j += blockDim.x) {
      float4 s = s4[j], xv = x4[j], o;
      o.x = (s.x >= thresh ? 1.0f : 0.0f) * xv.x;
      o.y = (s.y >= thresh ? 1.0f : 0.0f) * xv.y;
      o.z = (s.z >= thresh ? 1.0f : 0.0f) * xv.z;
      o.w = (s.w >= thresh ? 1.0f : 0.0f) * xv.w;
      o4[j] = o;
    }
    __syncthreads();  // buffers free before wave 0 reloads them
  }
#else
  long long n4 = tiles_total * (TILE / 4);
  long long stride = (long long)gridDim.x * blockDim.x;
  for (long long i = (long long)blockIdx.x * blockDim.x + threadIdx.x; i < n4; i += stride) {
    float4 s = ((const float4*)scores)[i];
    float4 xv = ((const float4*)x)[i];
    float4 o;
    o.x = (s.x >= thresh ? 1.0f : 0.0f) * xv.x;
    o.y = (s.y >= thresh ? 1.0f : 0.0f) * xv.y;
    o.z = (s.z >= thresh ? 1.0f : 0.0f) * xv.z;
    o.w = (s.w >= thresh ? 1.0f : 0.0f) * xv.w;
    ((float4*)out)[i] = o;
  }
#endif
}

__global__ void copy_kernel(const float* __restrict__ src, float* __restrict__ dst, int n) {
  int i = blockIdx.x * blockDim.x + threadIdx.x;
  if (i < n) dst[i] = src[i];
}

extern "C" void kernel_launch(void* const* d_in, const int* in_sizes, int n_in,
                              void* d_out, int out_size, void* d_ws, size_t ws_size,
                              hipStream_t stream) {
  const float* x      = (const float*)d_in[0];
  const float* scores = (const float*)d_in[1];
  const float* bias   = (const float*)d_in[2];
  float* out = (float*)d_out;
  long long n = in_sizes[0];
  int nb = in_sizes[2];
  uint32 k = (uint32)((double)n * 0.5);  // matches int(n * (1 - SPARSITY))

  uint32* ws    = (uint32*)d_ws;
  uint32* hist0 = ws;
  uint32* hist1 = ws + 4096;
  uint32* hist2 = ws + 8192;
  uint32* state = ws + STATE_OFF;

  init_ws<<<(WS_WORDS + 255) / 256, 256, 0, stream>>>(ws, k);

  long long n4 = n / 4;
  hist_kernel<0><<<HIST_BLOCKS, 256, 0, stream>>>((const float4*)scores, n4, hist0, state);
  scan_kernel<<<1, SCAN_THREADS, 0, stream>>>(hist0, state, 4096, 12, 0);
  hist_kernel<1><<<HIST_BLOCKS, 256, 0, stream>>>((const float4*)scores, n4, hist1, state);
  scan_kernel<<<1, SCAN_THREADS, 0, stream>>>(hist1, state, 4096, 12, 0);
  hist_kernel<2><<<HIST_BLOCKS, 256, 0, stream>>>((const float4*)scores, n4, hist2, state);
  scan_kernel<<<1, SCAN_THREADS, 0, stream>>>(hist2, state, 256, 8, 1);

  long long tiles = n / TILE;       // n = 33,554,432 -> 8192 tiles
  int tpb = 8;
  int blocks = (int)((tiles + tpb - 1) / tpb);
  mask_kernel<<<blocks, 256, 0, stream>>>(scores, x, out, state, tiles, tpb);
  copy_kernel<<<(nb + 255) / 256, 256, 0, stream>>>(bias, out + n, nb);
}